// ConditionalAttention_52888227283515
// MI455X (gfx1250) — compile-verified
//
#include <hip/hip_runtime.h>
#include <hip/hip_bf16.h>

typedef __attribute__((ext_vector_type(2))) float v2f;
typedef __attribute__((ext_vector_type(8))) float v8f;

// Vector-of-4-int types matching the async-LDS builtin signature
// (clang encoding: vV4i*1 V4i*3 Ii Ii).
typedef int avec4 __attribute__((vector_size(16)));
typedef __attribute__((address_space(1))) avec4 gvec4;
typedef __attribute__((address_space(3))) avec4 lvec4;

#if __has_builtin(__builtin_amdgcn_global_load_async_to_lds_b128) && \
    __has_builtin(__builtin_amdgcn_s_wait_asynccnt)
#define HAS_ASYNC_LDS 1
#else
#define HAS_ASYNC_LDS 0
#endif

// ---------------------------------------------------------------------------
// Generic fp32 WMMA GEMM: C[M,Nout] = A[M,K] @ W[Nout,K]^T + bias (+ addsrc) (relu?)
// Uses V_WMMA_F32_16X16X4_F32 (wave32). All M multiples of 16, Nout multiples
// of 128, K multiples of 8 in this workload.
// Block = 256 threads = 8 waves; each wave owns one 16x16 output tile,
// block covers 16 rows x 128 cols. A tile staged in LDS via async
// global->LDS (B128, ASYNCcnt) when available; padded row stride K+4 floats
// keeps 16B alignment for B128 LDS writes AND bank-conflict-free reads
// (banks (4*r + c) mod 64 distinct for r=0..15).
// Two independent accumulators break the WMMA RAW chain.
// ---------------------------------------------------------------------------
__global__ __launch_bounds__(256) void gemm_wmma_f32(
    const float* __restrict__ A, const float* __restrict__ W,
    const float* __restrict__ bias, const float* __restrict__ addsrc,
    float* __restrict__ C, int M, int Nout, int K, int relu) {
  __shared__ __align__(16) float As[16 * 260];  // supports K up to 256 (padded +4)
  const int Kp = K + 4;
  const int tid = threadIdx.x;
  const int mbase = blockIdx.y * 16;

#if HAS_ASYNC_LDS
  // Async staging: each thread moves 16-byte chunks global -> LDS directly.
  {
    const int kchunks = K >> 2;        // 16B chunks per row
    const int nchunks = 16 * kchunks;  // total chunks in the 16 x K tile
    for (int cidx = tid; cidx < nchunks; cidx += 256) {
      int r = cidx / kchunks;
      int kc = cidx - r * kchunks;
      const float* gsrc = A + (size_t)(mbase + r) * K + (kc << 2);
      float* lp = &As[r * Kp + (kc << 2)];
      __builtin_amdgcn_global_load_async_to_lds_b128(
          (gvec4*)gsrc, (lvec4*)lp, /*offset=*/0, /*cpol=*/0);
    }
    __builtin_amdgcn_s_wait_asynccnt(0);
  }
  __syncthreads();
#else
  // Fallback: cooperative load of the 16 x K A tile (coalesced along K)
  for (int idx = tid; idx < 16 * K; idx += 256) {
    int r = idx / K;
    int k = idx - r * K;
    As[r * Kp + k] = A[(size_t)(mbase + r) * K + k];
  }
  __syncthreads();
#endif

  const int wave = tid >> 5;
  const int lane = tid & 31;
  const int nbase = blockIdx.x * 128 + wave * 16;
  if (nbase >= Nout) return;

  const int half = lane >> 4;   // 0: K=kk+0/1, 1: K=kk+2/3
  const int l16  = lane & 15;   // A row / B col within tile
  const int koff = half * 2;

  const float* arow = As + l16 * Kp + koff;
  const float* wrow = W + (size_t)(nbase + l16) * K + koff;

  v8f acc0 = {};
  v8f acc1 = {};
  for (int kk = 0; kk < K; kk += 8) {
    v2f a0, b0, a1, b1;
    a0.x = arow[kk];     a0.y = arow[kk + 1];
    b0.x = wrow[kk];     b0.y = wrow[kk + 1];
    a1.x = arow[kk + 4]; a1.y = arow[kk + 5];
    b1.x = wrow[kk + 4]; b1.y = wrow[kk + 5];
    acc0 = __builtin_amdgcn_wmma_f32_16x16x4_f32(
        false, a0, false, b0, (short)0, acc0, false, false);
    acc1 = __builtin_amdgcn_wmma_f32_16x16x4_f32(
        false, a1, false, b1, (short)0, acc1, false, false);
  }
  v8f acc = acc0 + acc1;

  const int ncol = nbase + l16;
  const float bv = bias[ncol];
#pragma unroll
  for (int i = 0; i < 8; ++i) {
    int mrow = mbase + i + half * 8;
    size_t oidx = (size_t)mrow * Nout + ncol;
    float v = acc[i] + bv;
    if (addsrc) v += addsrc[oidx];
    if (relu) v = fmaxf(v, 0.0f);
    C[oidx] = v;
  }
}

// ---------------------------------------------------------------------------
// Edge pass A: c = relu(sign((Q[dst]+K[src])*Ew)*sqrt|.| + Eb)
//              score_h = clip(sum_d c*wscore, -5, 5); escore = exp(score)
//              den[dst,h] += escore   (segment-softmax denominator)
// 256 threads = 2 edges x 128 channels.
// ---------------------------------------------------------------------------
__global__ void edge_pass_a(const float* __restrict__ qkv,
                            const float* __restrict__ Eh,
                            const int* __restrict__ ei,
                            const float* __restrict__ wscore,
                            float* __restrict__ cbuf,
                            float* __restrict__ escore,
                            float* __restrict__ den, int E) {
  int t = threadIdx.x;
  int e = blockIdx.x * 2 + (t >> 7);
  if (e >= E) return;
  int j = t & 127;
  int dst = ei[e];
  int src = ei[E + e];

  float q  = qkv[(size_t)dst * 384 + j];
  float k  = qkv[(size_t)src * 384 + 128 + j];
  float ew = Eh[(size_t)e * 256 + j];
  float eb = Eh[(size_t)e * 256 + 128 + j];

  float tt = (q + k) * ew;
  float c = copysignf(sqrtf(fabsf(tt)), tt) + eb;
  c = fmaxf(c, 0.0f);
  cbuf[(size_t)e * 128 + j] = c;

  // score: sum over dh=16 channels of a head; wscore is (dh=16, h=8, 1)
  float sc = c * wscore[(j & 15) * 8 + (j >> 4)];
  sc += __shfl_xor(sc, 8, 16);
  sc += __shfl_xor(sc, 4, 16);
  sc += __shfl_xor(sc, 2, 16);
  sc += __shfl_xor(sc, 1, 16);
  if ((j & 15) == 0) {
    int h = j >> 4;
    float s = fminf(fmaxf(sc, -5.0f), 5.0f);
    float es = expf(s);  // s in [-5,5] -> no max-subtraction needed
    escore[(size_t)e * 8 + h] = es;
    atomicAdd(&den[(size_t)dst * 8 + h], es);
  }
}

// ---------------------------------------------------------------------------
// Edge pass B: alpha = escore/(den[dst]+1e-16); c2 = relu(bn(yc2)) -> out;
//              agg[dst] += (V[src] + c2) * alpha   (= nagg + cagg fused)
// ---------------------------------------------------------------------------
__global__ void edge_pass_c(const float* __restrict__ qkv,
                            const float* __restrict__ yc2,
                            const int* __restrict__ ei,
                            const float* __restrict__ escore,
                            const float* __restrict__ den,
                            const float* __restrict__ ss,
                            float* __restrict__ out_c2,
                            float* __restrict__ agg, int E) {
  int t = threadIdx.x;
  int e = blockIdx.x * 2 + (t >> 7);
  if (e >= E) return;
  int j = t & 127;
  int dst = ei[e];
  int src = ei[E + e];
  int h = j >> 4;

  float alpha = escore[(size_t)e * 8 + h] / (den[(size_t)dst * 8 + h] + 1e-16f);
  float y = yc2[(size_t)e * 128 + j];
  float c2 = fmaxf(y * ss[j] + ss[128 + j], 0.0f);
  out_c2[(size_t)e * 128 + j] = c2;

  float v = qkv[(size_t)src * 384 + 256 + j];
  atomicAdd(&agg[(size_t)dst * 128 + j], (v + c2) * alpha);
}

// ---------------------------------------------------------------------------
// Node: degree scaling + residual:  hres = nh*dc0 + nh*sqrt_deg*dc1 + x
// ---------------------------------------------------------------------------
__global__ void node_deg(const float* __restrict__ agg,
                         const float* __restrict__ sqrt_deg,
                         const float* __restrict__ dc,
                         const float* __restrict__ x,
                         float* __restrict__ hres) {
  int i = blockIdx.x;
  int j = threadIdx.x;
  float nh = agg[(size_t)i * 128 + j];
  float sd = sqrt_deg[i];
  nh = nh * dc[2 * j] + nh * sd * dc[2 * j + 1];
  hres[(size_t)i * 128 + j] = nh + x[(size_t)i * 128 + j];
}

// ---------------------------------------------------------------------------
// Batch-norm channel statistics (sum, sumsq) over M rows of 128 channels.
// ---------------------------------------------------------------------------
__global__ void bn_stats(const float* __restrict__ Y, int M,
                         float* __restrict__ stats) {
  int ch = threadIdx.x;
  float s = 0.0f, q = 0.0f;
  for (int r = blockIdx.x; r < M; r += gridDim.x) {
    float v = Y[(size_t)r * 128 + ch];
    s += v;
    q += v * v;
  }
  atomicAdd(&stats[ch], s);
  atomicAdd(&stats[128 + ch], q);
}

__global__ void bn_finalize(const float* __restrict__ stats,
                            const float* __restrict__ g,
                            const float* __restrict__ b, int M,
                            float* __restrict__ ss) {
  int ch = threadIdx.x;
  float inv = 1.0f / (float)M;
  float mean = stats[ch] * inv;
  float var = stats[128 + ch] * inv - mean * mean;
  float sc = g[ch] * rsqrtf(var + 1e-5f);
  ss[ch] = sc;
  ss[128 + ch] = b[ch] - mean * sc;
}

__global__ void apply_bn(const float* __restrict__ Y,
                         const float* __restrict__ ss,
                         float* __restrict__ out, size_t total) {
  size_t i = (size_t)blockIdx.x * blockDim.x + threadIdx.x;
  if (i >= total) return;
  int ch = (int)(i & 127);
  out[i] = Y[i] * ss[ch] + ss[128 + ch];
}

// ---------------------------------------------------------------------------
extern "C" void kernel_launch(void* const* d_in, const int* in_sizes, int n_in,
                              void* d_out, int out_size, void* d_ws,
                              size_t ws_size, hipStream_t stream) {
  const float* x        = (const float*)d_in[0];
  const float* conn_ex  = (const float*)d_in[1];
  const int*   ei       = (const int*)d_in[2];
  const float* sqrt_deg = (const float*)d_in[3];
  const float* qkv_w    = (const float*)d_in[4];
  const float* qkv_b    = (const float*)d_in[5];
  const float* w1       = (const float*)d_in[6];
  const float* b1       = (const float*)d_in[7];
  const float* wscore   = (const float*)d_in[8];
  const float* w2       = (const float*)d_in[9];
  const float* b2       = (const float*)d_in[10];
  const float* bn_c_g   = (const float*)d_in[11];
  const float* bn_c_b   = (const float*)d_in[12];
  const float* deg_coef = (const float*)d_in[13];
  const float* ffn1_w   = (const float*)d_in[14];
  const float* ffn1_b   = (const float*)d_in[15];
  const float* ffn2_w   = (const float*)d_in[16];
  const float* ffn2_b   = (const float*)d_in[17];
  const float* bn1_g    = (const float*)d_in[18];
  const float* bn1_b    = (const float*)d_in[19];
  const float* bn2_g    = (const float*)d_in[20];
  const float* bn2_b    = (const float*)d_in[21];

  const int N = in_sizes[0] / 128;  // 10000
  const int E = in_sizes[1] / 128;  // 160000

  float* out_nh = (float*)d_out;                   // (N,128)
  float* out_c2 = (float*)d_out + (size_t)N * 128; // (E,128)

  // Workspace carve-up (floats)
  float* ws = (float*)d_ws;
  size_t off = 0;
  float* qkv    = ws + off; off += (size_t)N * 384;
  float* Ehb    = ws + off; off += (size_t)E * 256;
  float* cbuf   = ws + off; off += (size_t)E * 128;
  float* escore = ws + off; off += (size_t)E * 8;
  float* den    = ws + off; off += (size_t)N * 8;
  float* yc2    = ws + off; off += (size_t)E * 128;
  float* agg    = ws + off; off += (size_t)N * 128;
  float* hres   = ws + off; off += (size_t)N * 128;
  float* xb     = ws + off; off += (size_t)N * 128;
  float* tbuf   = ws + off; off += (size_t)N * 256;
  float* zbuf   = ws + off; off += (size_t)N * 128;
  float* statsc = ws + off; off += 256;  // keep the 3 stats blocks contiguous
  float* stats1 = ws + off; off += 256;
  float* stats2 = ws + off; off += 256;
  float* ssc    = ws + off; off += 256;
  float* ss1    = ws + off; off += 256;
  float* ss2    = ws + off; off += 256;

  // Zero the accumulators used with atomics (graph-capturable memsets)
  (void)hipMemsetAsync(den, 0, (size_t)N * 8 * sizeof(float), stream);
  (void)hipMemsetAsync(agg, 0, (size_t)N * 128 * sizeof(float), stream);
  (void)hipMemsetAsync(statsc, 0, 3 * 256 * sizeof(float), stream);

  // 1) qkv = x @ qkv_w^T + qkv_b            (N x 384, K=128)
  gemm_wmma_f32<<<dim3(3, N / 16), 256, 0, stream>>>(
      x, qkv_w, qkv_b, nullptr, qkv, N, 384, 128, 0);

  // 2) Eh = conn_ex @ w1^T + b1             (E x 256, K=128)
  gemm_wmma_f32<<<dim3(2, E / 16), 256, 0, stream>>>(
      conn_ex, w1, b1, nullptr, Ehb, E, 256, 128, 0);

  // 3) edge gate + scores + softmax denominators
  edge_pass_a<<<(E + 1) / 2, 256, 0, stream>>>(qkv, Ehb, ei, wscore, cbuf,
                                               escore, den, E);

  // 4) yc2 = c @ w2^T + b2 + conn_ex        (E x 128, K=128)
  gemm_wmma_f32<<<dim3(1, E / 16), 256, 0, stream>>>(
      cbuf, w2, b2, conn_ex, yc2, E, 128, 128, 0);

  // 5) BN_c statistics + finalize
  bn_stats<<<1024, 128, 0, stream>>>(yc2, E, statsc);
  bn_finalize<<<1, 128, 0, stream>>>(statsc, bn_c_g, bn_c_b, E, ssc);

  // 6) alpha, c2 output, fused (V + c2)*alpha scatter
  edge_pass_c<<<(E + 1) / 2, 256, 0, stream>>>(qkv, yc2, ei, escore, den, ssc,
                                               out_c2, agg, E);

  // 7) degree scaling + residual
  node_deg<<<N, 128, 0, stream>>>(agg, sqrt_deg, deg_coef, x, hres);

  // 8) BN1 -> xb
  bn_stats<<<1024, 128, 0, stream>>>(hres, N, stats1);
  bn_finalize<<<1, 128, 0, stream>>>(stats1, bn1_g, bn1_b, N, ss1);
  {
    size_t total = (size_t)N * 128;
    apply_bn<<<(unsigned)((total + 255) / 256), 256, 0, stream>>>(hres, ss1, xb,
                                                                  total);
  }

  // 9) FFN: t = relu(xb @ ffn1_w^T + b)     (N x 256, K=128)
  gemm_wmma_f32<<<dim3(2, N / 16), 256, 0, stream>>>(
      xb, ffn1_w, ffn1_b, nullptr, tbuf, N, 256, 128, 1);

  // 10) z = t @ ffn2_w^T + b + hres         (N x 128, K=256)
  gemm_wmma_f32<<<dim3(1, N / 16), 256, 0, stream>>>(
      tbuf, ffn2_w, ffn2_b, hres, zbuf, N, 128, 256, 0);

  // 11) BN2 -> out_nh
  bn_stats<<<1024, 128, 0, stream>>>(zbuf, N, stats2);
  bn_finalize<<<1, 128, 0, stream>>>(stats2, bn2_g, bn2_b, N, ss2);
  {
    size_t total = (size_t)N * 128;
    apply_bn<<<(unsigned)((total + 255) / 256), 256, 0, stream>>>(zbuf, ss2,
                                                                  out_nh, total);
  }

  (void)n_in;
  (void)out_size;
  (void)ws_size;
}